// UnsupervisedPixelContrastLoss_85676007620955
// MI455X (gfx1250) — compile-verified
//
#include <hip/hip_runtime.h>
#include <hip/hip_bf16.h>

typedef __attribute__((ext_vector_type(2))) float v2f;
typedef __attribute__((ext_vector_type(8))) float v8f;
typedef __attribute__((address_space(3))) char lds_char;

#define N_ROWS   8192          // b*h*w = 2*64*64
#define DDIM     64
#define TEMP_INV 10.0f         // 1/temperature, folded into A operand
#define EPSN     1e-12f

#define N_TILES   (N_ROWS / 16)     // 512 16-row/col tiles
#define CSPLIT    4                 // column-quarter split
#define TILES_PER_WAVE (N_TILES / CSPLIT)   // 128 column tiles per wave
#define PAD_TILES 3                 // dummy B tiles so 3-ahead async issue needs no tail code

// Workspace layout (floats)
#define Z1A_OFF   0                                     // A operand, fragment layout
#define Z2B_OFF   (N_ROWS * DDIM)                       // B operand, tile-linear fragment layout (+pad)
#define DIAG_OFF  (2 * N_ROWS * DDIM + PAD_TILES * 1024)
#define PART_OFF  (DIAG_OFF + N_ROWS)                   // CSPLIT partial sum(exp) per row

// ---------------------------------------------------------------------------
// Kernel 1: normalize + transpose into WMMA f32 16x16x4 fragment layout + diag.
// Input layout (b, d, h, w): element (row n, dim d) at b*64*4096 + d*4096 + (n&4095).
// Fragment layout (ISA 7.12.2): lane l holds M(or N)=l%16, K=4*kk+2*(l/16)+{0,1};
// stored so one b64 per lane fetches a fragment:  frag[(tile*16+kk)*32 + lane] : v2f.
// B tiles are thus 4KB-contiguous per column tile (async-copy friendly).
// ---------------------------------------------------------------------------
__global__ __launch_bounds__(256) void pcl_normalize_kernel(
    const float* __restrict__ z1, const float* __restrict__ z2,
    float* __restrict__ ws) {
  __shared__ float t1[64][65];
  __shared__ float t2[64][65];
  __shared__ float inv1[64], inv2[64], dotv[64];

  const int tid     = threadIdx.x;
  const int rowbase = blockIdx.x * 64;
  const int b       = rowbase >> 12;
  const int hwbase  = rowbase & 4095;
  const float* p1 = z1 + b * (DDIM * 4096) + hwbase;
  const float* p2 = z2 + b * (DDIM * 4096) + hwbase;

  #pragma unroll
  for (int it = 0; it < 16; ++it) {
    int idx = it * 256 + tid;
    int d = idx >> 6, r = idx & 63;
    t1[d][r] = p1[d * 4096 + r];
    t2[d][r] = p2[d * 4096 + r];
  }
  __syncthreads();

  if (tid < 64) {
    float s = 0.f;
    #pragma unroll
    for (int d = 0; d < 64; ++d) { float v = t1[d][tid]; s = fmaf(v, v, s); }
    inv1[tid] = TEMP_INV / fmaxf(sqrtf(s), EPSN);
  } else if (tid < 128) {
    int r = tid - 64;
    float s = 0.f;
    #pragma unroll
    for (int d = 0; d < 64; ++d) { float v = t2[d][r]; s = fmaf(v, v, s); }
    inv2[r] = 1.0f / fmaxf(sqrtf(s), EPSN);
  } else if (tid < 192) {
    int r = tid - 128;
    float s = 0.f;
    #pragma unroll
    for (int d = 0; d < 64; ++d) s = fmaf(t1[d][r], t2[d][r], s);
    dotv[r] = s;
  }
  __syncthreads();

  if (tid < 64)
    ws[DIAG_OFF + rowbase + tid] = dotv[tid] * inv1[tid] * inv2[tid];

  v2f* z1a = (v2f*)(ws + Z1A_OFF);
  v2f* z2b = (v2f*)(ws + Z2B_OFF);
  const int l = tid & 31, g = tid >> 5;
  #pragma unroll
  for (int j = 0; j < 8; ++j) {
    int flat = g * 8 + j;
    int tl = flat >> 4, kk = flat & 15;
    int K0 = kk * 4 + ((l >> 4) << 1);
    int m  = tl * 16 + (l & 15);
    int itile = blockIdx.x * 4 + tl;
    v2f a;  a.x  = inv1[m] * t1[K0][m]; a.y  = inv1[m] * t1[K0 + 1][m];
    v2f bb; bb.x = inv2[m] * t2[K0][m]; bb.y = inv2[m] * t2[K0 + 1][m];
    z1a[(itile * 16 + kk) * 32 + l] = a;
    z2b[(itile * 16 + kk) * 32 + l] = bb;
  }
}

// ---------------------------------------------------------------------------
// Kernel 2: fused GEMM (f32 WMMA) + streaming sum(exp), B staged via async
// Global->LDS copies shared by the 8 waves of a block.
// Grid: 256 blocks = 64 row-tile-groups x 4 column-quarters; wave w owns
// row tile group*8+w and 128 column tiles.  4-deep LDS ring of 4KB B tiles,
// filled 3 ahead (per-lane 16B global_load_async_to_lds_b128), consumed via
// ds_load_b64.  sim entries are in [-10,10] -> plain sum(exp) is f32-safe.
// ---------------------------------------------------------------------------
__device__ __forceinline__ void async_tile_copy(unsigned lds_byte, unsigned gbyte,
                                                unsigned long long gbase) {
  // copies 16 bytes per lane from global address gbase plus offset gbyte to the LDS address in lds_byte (ASYNCcnt-tracked)
  asm volatile("global_load_async_to_lds_b128 %0, %1, %2"
               :: "v"(lds_byte), "v"(gbyte), "s"(gbase) : "memory");
}

__global__ __launch_bounds__(256) void pcl_gemm_lse_kernel(float* __restrict__ ws) {
  __shared__ __align__(16) char smem[4 * 4096];   // 4-deep ring of 4KB B tiles

  const v2f* z1a = (const v2f*)(ws + Z1A_OFF);
  const float* z2bf = ws + Z2B_OFF;
  float* part = ws + PART_OFF;

  const int tid  = threadIdx.x;
  const int lane = tid & 31;
  const int it   = (blockIdx.x >> 2) * 8 + (tid >> 5);  // row tile 0..511
  const int q    = blockIdx.x & 3;                      // column quarter
  const int jt0  = q * TILES_PER_WAVE;

  const unsigned lbase = (unsigned)(size_t)(lds_char*)&smem[0];
  const unsigned long long gbase = (unsigned long long)z2bf;

  // A fragments for this strip, resident in VGPRs for the whole kernel.
  v2f a[16];
  #pragma unroll
  for (int kk = 0; kk < 16; ++kk)
    a[kk] = z1a[(it * 16 + kk) * 32 + lane];

  // Prime the ring: tiles jt0..jt0+2 into buffers 0..2.
  #pragma unroll
  for (int p = 0; p < 3; ++p)
    async_tile_copy(lbase + (unsigned)p * 4096u + (unsigned)tid * 16u,
                    (unsigned)(jt0 + p) * 4096u + (unsigned)tid * 16u, gbase);

  v8f acc = {0.f, 0.f, 0.f, 0.f, 0.f, 0.f, 0.f, 0.f};

  for (int j = 0; j < TILES_PER_WAVE; ++j) {
    // Oldest outstanding copy (tile j) done when <=2 remain (in-order dones).
    asm volatile("s_wait_asynccnt 0x2" ::: "memory");
    __syncthreads();   // all waves' portions of tile j landed; buf[(j+3)&3] reads retired

    // Issue tile j+3 (Z2B is padded by 3 dummy tiles -> no tail branch).
    async_tile_copy(lbase + (unsigned)((j + 3) & 3) * 4096u + (unsigned)tid * 16u,
                    (unsigned)(jt0 + j + 3) * 4096u + (unsigned)tid * 16u, gbase);

    const v2f* bufv = (const v2f*)(smem + (j & 3) * 4096);
    v2f bf[16];
    #pragma unroll
    for (int kk = 0; kk < 16; ++kk)        // ds_load_b64, conflict-free
      bf[kk] = bufv[kk * 32 + lane];

    v8f c = {0.f, 0.f, 0.f, 0.f, 0.f, 0.f, 0.f, 0.f};
    #pragma unroll
    for (int kk = 0; kk < 16; ++kk)
      c = __builtin_amdgcn_wmma_f32_16x16x4_f32(
              false, a[kk], false, bf[kk], (short)0, c, false, false);

    #pragma unroll
    for (int v = 0; v < 8; ++v)
      acc[v] += __expf(c[v]);
  }

  // C layout: VGPR v = row M=v (lanes 0-15) / M=v+8 (lanes 16-31), col = lane%16.
  #pragma unroll
  for (int v = 0; v < 8; ++v) {
    float s = acc[v];
    s += __shfl_xor(s, 1, 32);
    s += __shfl_xor(s, 2, 32);
    s += __shfl_xor(s, 4, 32);
    s += __shfl_xor(s, 8, 32);
    if ((lane & 15) == 0) {
      int m = v + ((lane >> 4) << 3);
      part[q * N_ROWS + it * 16 + m] = s;   // partial sum(exp) for this quarter
    }
  }
}

// ---------------------------------------------------------------------------
// Kernel 3: combine quarters, lse - diag, deterministic mean (no atomics).
// ---------------------------------------------------------------------------
__global__ __launch_bounds__(256) void pcl_reduce_kernel(const float* __restrict__ ws,
                                                         float* __restrict__ out) {
  __shared__ float sh[256];
  const int tid = threadIdx.x;
  const float* part = ws + PART_OFF;
  const float* diag = ws + DIAG_OFF;
  float s = 0.f;
  #pragma unroll
  for (int i = 0; i < 32; ++i) {
    int row = tid + i * 256;
    float p = part[row] + part[N_ROWS + row] +
              part[2 * N_ROWS + row] + part[3 * N_ROWS + row];
    s += __logf(p) - diag[row];
  }
  sh[tid] = s;
  __syncthreads();
  for (int off = 128; off > 0; off >>= 1) {
    if (tid < off) sh[tid] += sh[tid + off];
    __syncthreads();
  }
  if (tid == 0) out[0] = sh[0] * (1.0f / (float)N_ROWS);
}

extern "C" void kernel_launch(void* const* d_in, const int* in_sizes, int n_in,
                              void* d_out, int out_size, void* d_ws, size_t ws_size,
                              hipStream_t stream) {
  const float* z1 = (const float*)d_in[0];
  const float* z2 = (const float*)d_in[1];
  float* ws  = (float*)d_ws;
  float* out = (float*)d_out;

  pcl_normalize_kernel<<<N_ROWS / 64, 256, 0, stream>>>(z1, z2, ws);
  pcl_gemm_lse_kernel<<<(N_TILES / 8) * CSPLIT, 256, 0, stream>>>(ws);
  pcl_reduce_kernel<<<1, 256, 0, stream>>>(ws, out);
}